// RLnetwork_24919400251949
// MI455X (gfx1250) — compile-verified
//
#include <hip/hip_runtime.h>

// ---------------------------------------------------------------------------
// RL structure2vec network on gfx1250 (MI455X). fp32 WMMA (16x16x4) path.
// N=2000, E=64, T=4.
// ---------------------------------------------------------------------------

typedef float v2f __attribute__((ext_vector_type(2)));
typedef float v8f __attribute__((ext_vector_type(8)));

#define NN    2000
#define EE    64
#define NE    (NN * EE)          // 128000

// ---------------------------------------------------------------------------
// s3pre[j][e] = sum_i relu(W[i*N + j] * t4w[e] + t4b[e])
// block: 256 threads = 4 columns x 64 features; grid = 500.
// Lanes within a wave share j => W reads are wave-broadcast; the 4 columns of
// a block are adjacent so the cacheline is well used across blocks.
// ---------------------------------------------------------------------------
__global__ void s3pre_kernel(const float* __restrict__ W,
                             const float* __restrict__ t4w,
                             const float* __restrict__ t4b,
                             float* __restrict__ s3pre) {
  const int tid = threadIdx.x;
  const int e = tid & 63;
  const int j = blockIdx.x * 4 + (tid >> 6);
  const float a = t4w[e];
  const float b = t4b[e];
  float acc = 0.0f;
  for (int i = 0; i < NN; ++i) {
    float w = W[i * NN + j];
    acc += fmaxf(fmaf(w, a, b), 0.0f);
  }
  s3pre[j * EE + e] = acc;
}

// ---------------------------------------------------------------------------
// out[M x 64] = A[M x 64] @ B[64 x 64] + bias, optional relu.
// block = 128 threads = 4 waves; block owns a 16-row M tile; wave w owns the
// 16-wide N tile starting at 16*w. 16 chained v_wmma_f32_16x16x4_f32 per wave.
// M must be a multiple of 16 (2000 = 125 * 16). Grid = M/16.
// ---------------------------------------------------------------------------
__global__ void gemm64_kernel(const float* __restrict__ A,
                              const float* __restrict__ B,
                              const float* __restrict__ bias,
                              float* __restrict__ out,
                              int do_relu) {
  __shared__ float lA[16 * 65];   // padded stride 65 -> conflict-free frag reads
  __shared__ float lB[64 * 64];
  const int tid = threadIdx.x;          // 0..127
  const int m0 = blockIdx.x * 16;

  for (int i = tid; i < 16 * 64; i += 128) {
    int r = i >> 6, c = i & 63;
    lA[r * 65 + c] = A[(m0 + r) * EE + c];
  }
  for (int i = tid; i < 64 * 64; i += 128) lB[i] = B[i];
  __syncthreads();

  const int wave = tid >> 5;
  const int lane = tid & 31;
  const int half = lane >> 4;           // 0: K 0/1 slice, 1: K 2/3 slice
  const int r    = lane & 15;           // A row / B column within tile
  const int e0   = wave * 16;

  v8f acc = {};
#pragma unroll
  for (int k = 0; k < 64; k += 4) {
    v2f a, b;
    a.x = lA[r * 65 + k + 2 * half + 0];
    a.y = lA[r * 65 + k + 2 * half + 1];
    b.x = lB[(k + 2 * half + 0) * 64 + e0 + r];
    b.y = lB[(k + 2 * half + 1) * 64 + e0 + r];
    acc = __builtin_amdgcn_wmma_f32_16x16x4_f32(false, a, false, b,
                                                (short)0, acc, false, false);
  }

  const float bv = bias[e0 + r];
#pragma unroll
  for (int v = 0; v < 8; ++v) {
    float val = acc[v] + bv;
    if (do_relu) val = fmaxf(val, 0.0f);
    out[(m0 + v + 8 * half) * EE + e0 + r] = val;   // D row = v + 8*half
  }
}

// ---------------------------------------------------------------------------
// h[N x 64] = (W > 0) @ mu[N x 64].  adj built on the fly while staging into
// LDS (saves 16 MB of workspace; W stays resident in the 192 MB L2 anyway).
// block = 128 threads = 4 waves; 16-row M tile; K chunked by 80 (25 chunks).
// 500 f32 WMMAs per wave.
// ---------------------------------------------------------------------------
#define KC 80
__global__ void adjmm_kernel(const float* __restrict__ W,
                             const float* __restrict__ mu,
                             float* __restrict__ out) {
  __shared__ float lA[16 * (KC + 1)];
  __shared__ float lB[KC * 64];
  const int tid = threadIdx.x;
  const int m0 = blockIdx.x * 16;
  const int wave = tid >> 5;
  const int lane = tid & 31;
  const int half = lane >> 4;
  const int r    = lane & 15;
  const int e0   = wave * 16;

  v8f acc = {};
  for (int k0 = 0; k0 < NN; k0 += KC) {
    __syncthreads();
    for (int i = tid; i < 16 * KC; i += 128) {
      int rr = i / KC, cc = i % KC;
      float w = W[(m0 + rr) * NN + (k0 + cc)];       // coalesced row reads
      lA[rr * (KC + 1) + cc] = (w > 0.0f) ? 1.0f : 0.0f;
    }
    for (int i = tid; i < KC * 64; i += 128) {
      int rr = i >> 6, cc = i & 63;
      lB[rr * 64 + cc] = mu[(k0 + rr) * EE + cc];
    }
    __syncthreads();
#pragma unroll 4
    for (int k = 0; k < KC; k += 4) {
      v2f a, b;
      a.x = lA[r * (KC + 1) + k + 2 * half + 0];
      a.y = lA[r * (KC + 1) + k + 2 * half + 1];
      b.x = lB[(k + 2 * half + 0) * 64 + e0 + r];
      b.y = lB[(k + 2 * half + 1) * 64 + e0 + r];
      acc = __builtin_amdgcn_wmma_f32_16x16x4_f32(false, a, false, b,
                                                  (short)0, acc, false, false);
    }
  }
#pragma unroll
  for (int v = 0; v < 8; ++v)
    out[(m0 + v + 8 * half) * EE + e0 + r] = acc[v];
}

// mu = relu(s1 + s2 + s3), elementwise over N*E
__global__ void mu_kernel(const float* __restrict__ s1,
                          const float* __restrict__ s2,
                          const float* __restrict__ s3,
                          float* __restrict__ mu) {
  int i = blockIdx.x * 256 + threadIdx.x;
  if (i < NE) mu[i] = fmaxf(s1[i] + s2[i] + s3[i], 0.0f);
}

// initial s2 = (adj @ 0) @ t2w + t2b = t2b broadcast
__global__ void init_s2_kernel(const float* __restrict__ t2b,
                               float* __restrict__ s2) {
  int i = blockIdx.x * 256 + threadIdx.x;
  if (i < NE) s2[i] = t2b[i & 63];
}

// c0 = t5b + sum_e relu( (colsum(mu) @ t6w + t6b)[e] ) * t5w[e]
// single workgroup of 64 threads; fixed summation order => deterministic.
__global__ void head_kernel(const float* __restrict__ mu,
                            const float* __restrict__ t6w,
                            const float* __restrict__ t6b,
                            const float* __restrict__ t5w,
                            const float* __restrict__ t5b,
                            float* __restrict__ c0) {
  __shared__ float cs[64];
  __shared__ float gv[64];
  const int e = threadIdx.x;            // 64 threads
  float s = 0.0f;
  for (int j = 0; j < NN; ++j) s += mu[j * EE + e];
  cs[e] = s;
  __syncthreads();
  float g = t6b[e];
  for (int k = 0; k < 64; ++k) g = fmaf(cs[k], t6w[k * 64 + e], g);
  gv[e] = fmaxf(g, 0.0f) * t5w[e];
  __syncthreads();
  if (e == 0) {
    float t = t5b[0];
    for (int k = 0; k < 64; ++k) t += gv[k];
    c0[0] = t;
  }
}

// est[j] = c0 + sum_e localRelu[j][e] * t5w[64 + e]
__global__ void final_kernel(const float* __restrict__ localRelu,
                             const float* __restrict__ t5w,
                             const float* __restrict__ c0,
                             float* __restrict__ out) {
  int j = blockIdx.x * 256 + threadIdx.x;
  if (j >= NN) return;
  float acc = c0[0];
  for (int e = 0; e < 64; ++e)
    acc = fmaf(localRelu[j * EE + e], t5w[64 + e], acc);
  out[j] = acc;
}

// ---------------------------------------------------------------------------
extern "C" void kernel_launch(void* const* d_in, const int* in_sizes, int n_in,
                              void* d_out, int out_size, void* d_ws, size_t ws_size,
                              hipStream_t stream) {
  (void)in_sizes; (void)n_in; (void)out_size; (void)ws_size;
  const float* xv  = (const float*)d_in[0];
  const float* W   = (const float*)d_in[1];
  const float* t1w = (const float*)d_in[2];
  const float* t1b = (const float*)d_in[3];
  const float* t2w = (const float*)d_in[4];
  const float* t2b = (const float*)d_in[5];
  const float* t3w = (const float*)d_in[6];
  const float* t3b = (const float*)d_in[7];
  const float* t4w = (const float*)d_in[8];
  const float* t4b = (const float*)d_in[9];
  const float* t5w = (const float*)d_in[10];
  const float* t5b = (const float*)d_in[11];
  const float* t6w = (const float*)d_in[12];
  const float* t6b = (const float*)d_in[13];
  const float* t7w = (const float*)d_in[14];
  const float* t7b = (const float*)d_in[15];

  float* ws  = (float*)d_ws;            // ~3.6 MB of fp32 scratch
  float* S1  = ws + 0 * NE;
  float* S3P = ws + 1 * NE;
  float* S3  = ws + 2 * NE;
  float* S2  = ws + 3 * NE;
  float* MU  = ws + 4 * NE;
  float* H   = ws + 5 * NE;
  float* LOC = ws + 6 * NE;
  float* C0  = ws + 7 * NE;

  const int GEMM_GRID = NN / 16;        // 125
  const int EW_GRID   = (NE + 255) / 256;

  s3pre_kernel<<<NN / 4, 256, 0, stream>>>(W, t4w, t4b, S3P);
  gemm64_kernel<<<GEMM_GRID, 128, 0, stream>>>(xv,  t1w, t1b, S1, 0);
  gemm64_kernel<<<GEMM_GRID, 128, 0, stream>>>(S3P, t3w, t3b, S3, 0);
  init_s2_kernel<<<EW_GRID, 256, 0, stream>>>(t2b, S2);

  for (int t = 0; t < 4; ++t) {
    mu_kernel<<<EW_GRID, 256, 0, stream>>>(S1, S2, S3, MU);
    if (t < 3) {                        // last-iteration s2 is dead in the ref
      adjmm_kernel<<<GEMM_GRID, 128, 0, stream>>>(W, MU, H);
      gemm64_kernel<<<GEMM_GRID, 128, 0, stream>>>(H, t2w, t2b, S2, 0);
    }
  }

  gemm64_kernel<<<GEMM_GRID, 128, 0, stream>>>(MU, t7w, t7b, LOC, 1);
  head_kernel<<<1, 64, 0, stream>>>(MU, t6w, t6b, t5w, t5b, C0);
  final_kernel<<<(NN + 255) / 256, 256, 0, stream>>>(LOC, t5w, C0, (float*)d_out);
}